// BM3D_90091234001110
// MI455X (gfx1250) — compile-verified
//
#include <hip/hip_runtime.h>

// ---------------------------------------------------------------------------
// BM3D-deblurring (RI filter + global empirical Wiener), fully on-device.
//   ri  = conj(H) / (|H|^2 + sigma^2)          (H = OTF of circularly-shifted PSF)
//   psd = sigma^2 * |ri|^2 * n
//   Z   = fft2(y) * ri
//   S   = max(|Z|^2/n - psd, 0);  W = S/(S+psd+eps)
//   out = real(ifft2(Z*W))
// Bandwidth-bound: ~1 GB of traffic @ 23.3 TB/s.  FFTs are LDS Stockham radix-2
// with a precomputed 1024-entry twiddle table (no transcendentals in hot loops).
// The OTF build is a genuine f32 GEMM (1024x1024x32) -> V_WMMA_F32_16X16X4_F32.
// ---------------------------------------------------------------------------

#define TWO_PI 6.283185307179586f
#define SIG2   0.0025f          /* SIGMA_PSD^2 = 0.05^2 */
#define EPSW   1e-12f
#define INV_N  (1.0f / 1048576.0f)

typedef __attribute__((ext_vector_type(2))) float v2f;
typedef __attribute__((ext_vector_type(8))) float v8f;

__device__ __forceinline__ float2 cmul(float2 a, float2 b) {
    return make_float2(a.x * b.x - a.y * b.y, a.x * b.y + a.y * b.x);
}

// ---------------------------------------------------------------------------
// 0) Twiddle table: tw[k] = exp(-2*pi*i*k/1024), k = 0..1023.
//    All FFT-stage twiddles and all PSF/OTF phase factors are exact lookups.
// ---------------------------------------------------------------------------
__global__ __launch_bounds__(256) void tw_kernel(float2* __restrict__ tw) {
    int k = blockIdx.x * 256 + threadIdx.x;
    if (k >= 1024) return;
    float sn, cs;
    sincosf(-TWO_PI * (float)k * (1.0f / 1024.0f), &sn, &cs);
    tw[k] = make_float2(cs, sn);
}

// One Stockham radix-2 butterfly for a 1024-point FFT living in LDS.
// Stage s: m = 2^s, l = 512>>s. Butterfly t in [0,512):
//   j = t >> s;  src a[t], a[t+512];  dst b[t + j*m], b[t + j*m + m]
//   w = exp(-+2*pi*i * j / (1024>>s)) = tw[j<<s] (fwd, dir=+1) / conj (inv, dir=-1)
__device__ __forceinline__ void fft_stage(const float2* a, float2* b,
                                          const float2* __restrict__ tw,
                                          int t, int s, float dir) {
    int m  = 1 << s;
    int j  = t >> s;
    float2 c0 = a[t];
    float2 c1 = a[t + 512];
    float2 w  = tw[j << s];
    float  wy = dir * w.y;
    float2 d  = make_float2(c0.x - c1.x, c0.y - c1.y);
    int d0 = t + (j << s);
    b[d0]     = make_float2(c0.x + c1.x, c0.y + c1.y);
    b[d0 + m] = make_float2(w.x * d.x - wy * d.y, w.x * d.y + wy * d.x);
}

// ---------------------------------------------------------------------------
// 1) G[i][v] = sum_j k[i,j] * exp(-2*pi*i * v*(j-12)/1024),  i in [0,32) (25..31 zero)
// ---------------------------------------------------------------------------
__global__ __launch_bounds__(256) void psf_g_kernel(const float* __restrict__ psf,
                                                    const float2* __restrict__ tw,
                                                    float* __restrict__ Gre,
                                                    float* __restrict__ Gim) {
    int idx = blockIdx.x * 256 + threadIdx.x;      // 32*1024 threads
    if (idx >= 32 * 1024) return;
    int i = idx >> 10, v = idx & 1023;
    float re = 0.f, im = 0.f;
    if (i < 25) {
        for (int j = 0; j < 25; ++j) {
            int p = (v * (j - 12)) & 1023;         // exact mod-1024 phase
            float2 w = tw[p];                      // e^{-2*pi*i*p/1024}
            float kv = psf[i * 25 + j];
            re += kv * w.x;
            im += kv * w.y;
        }
    }
    Gre[idx] = re;
    Gim[idx] = im;
}

// ---------------------------------------------------------------------------
// 2) Hf = F x G complex GEMM via V_WMMA_F32_16X16X4_F32, then ri/psd.
//    F[u][i] = exp(-i*theta), theta = 2*pi*u*(i-12)/1024  => c - i*s
//    Hre = sum c*Gre + s*Gim ;  Him = sum c*Gim - s*Gre
//    A (16x4 f32) layout per ISA: lanes 0-15 = M rows; VGPR0:{K0|K2}, VGPR1:{K1|K3}.
//    B (4x16 f32) layout by symmetry with C: lanes = N cols, same K order.
//    C (16x16 f32): VGPR g = row u0+g (lanes 0-15) / u0+g+8 (lanes 16-31).
// ---------------------------------------------------------------------------
__global__ __launch_bounds__(32) void otf_wmma_kernel(const float* __restrict__ Gre,
                                                      const float* __restrict__ Gim,
                                                      const float2* __restrict__ tw,
                                                      float2* __restrict__ ri,
                                                      float* __restrict__ psd) {
    int tile = blockIdx.x;                 // 64x64 tiles of 16x16
    int u0 = (tile >> 6) << 4;
    int v0 = (tile & 63) << 4;
    int l    = threadIdx.x;
    int half = l >> 4;
    int mn   = l & 15;                     // M row for A, N col for B/C
    int u = u0 + mn;
    int v = v0 + mn;

    v8f hre = 0.0f;
    v8f him = 0.0f;

#pragma unroll
    for (int kk = 0; kk < 32; kk += 4) {
        int i0 = kk + (half ? 2 : 0);      // K index held in vector elt 0
        int i1 = i0 + 1;                   // K index held in vector elt 1
        int p0 = (u * (i0 - 12)) & 1023;
        int p1 = (u * (i1 - 12)) & 1023;
        // e^{+i*theta} = conj(tw[p]):  c = tw.x, s = -tw.y
        float2 w0 = tw[p0];
        float2 w1 = tw[p1];
        v2f ac  = {w0.x, w1.x};
        v2f as  = {-w0.y, -w1.y};
        v2f asn = {w0.y, w1.y};
        v2f bre = {Gre[i0 * 1024 + v], Gre[i1 * 1024 + v]};
        v2f bim = {Gim[i0 * 1024 + v], Gim[i1 * 1024 + v]};
        hre = __builtin_amdgcn_wmma_f32_16x16x4_f32(false, ac,  false, bre, (short)0, hre, false, false);
        hre = __builtin_amdgcn_wmma_f32_16x16x4_f32(false, as,  false, bim, (short)0, hre, false, false);
        him = __builtin_amdgcn_wmma_f32_16x16x4_f32(false, ac,  false, bim, (short)0, him, false, false);
        him = __builtin_amdgcn_wmma_f32_16x16x4_f32(false, asn, false, bre, (short)0, him, false, false);
    }

#pragma unroll
    for (int g = 0; g < 8; ++g) {
        int uu = u0 + g + 8 * half;
        float a = hre[g], b = him[g];
        float den = a * a + b * b + SIG2;
        float rr = a / den, ii = -b / den;
        ri[uu * 1024 + v]  = make_float2(rr, ii);
        psd[uu * 1024 + v] = SIG2 * (rr * rr + ii * ii) * 1048576.0f;
    }
}

// ---------------------------------------------------------------------------
// 3) Forward 1024-pt FFT per row (real input -> complex spectrum row).
// ---------------------------------------------------------------------------
__global__ __launch_bounds__(256) void rowfft_fwd(const float* __restrict__ y,
                                                  const float2* __restrict__ tw,
                                                  float2* __restrict__ Z) {
    __shared__ float2 sA[1024];
    __shared__ float2 sB[1024];
    int row = blockIdx.x;                  // img*1024 + r (chunk-local)
    int tid = threadIdx.x;
    const float* src = y + (size_t)row * 1024;
    for (int i = tid; i < 1024; i += 256) sA[i] = make_float2(src[i], 0.f);
    __syncthreads();
    float2 *a = sA, *b = sB;
    for (int s = 0; s < 10; ++s) {
        fft_stage(a, b, tw, tid, s, 1.0f);
        fft_stage(a, b, tw, tid + 256, s, 1.0f);
        __syncthreads();
        float2* t = a; a = b; b = t;
    }
    float2* dst = Z + (size_t)row * 1024;
    for (int i = tid; i < 1024; i += 256) dst[i] = a[i];
}

// ---------------------------------------------------------------------------
// 4) Column pass, fused: forward col FFT -> *ri -> Wiener -> inverse col FFT.
//    One block = one image x 4 columns (LDS ping-pong 2 x 4 x 1024 x 8B = 64 KB).
// ---------------------------------------------------------------------------
__global__ __launch_bounds__(256) void colfft_wiener(float2* __restrict__ Z,
                                                     const float2* __restrict__ tw,
                                                     const float2* __restrict__ ri,
                                                     const float* __restrict__ psd) {
    __shared__ float2 sA[4096];            // [col*1024 + row]
    __shared__ float2 sB[4096];
    int blk = blockIdx.x;
    int img = blk >> 8;
    int c0  = (blk & 255) << 2;
    int tid = threadIdx.x;
    float2* base = Z + (size_t)img * 1048576;

    for (int li = tid; li < 4096; li += 256) {
        int r = li >> 2, c = li & 3;
        sA[c * 1024 + r] = base[(size_t)r * 1024 + c0 + c];
    }
    __syncthreads();

    float2 *a = sA, *b = sB;
    // forward FFT along each of the 4 columns
    for (int s = 0; s < 10; ++s) {
        for (int q = 0; q < 8; ++q) {
            int tt  = tid + (q << 8);
            int col = tt >> 9;
            int t   = tt & 511;
            fft_stage(a + (col << 10), b + (col << 10), tw, t, s, 1.0f);
        }
        __syncthreads();
        float2* t = a; a = b; b = t;
    }

    // pointwise: Z *= ri; empirical Wiener with colored-noise PSD
    for (int li = tid; li < 4096; li += 256) {
        int c = li >> 10, r = li & 1023;
        int uv = r * 1024 + (c0 + c);
        float2 z = cmul(a[li], ri[uv]);
        float p  = psd[uv];
        float S  = fmaxf((z.x * z.x + z.y * z.y) * INV_N - p, 0.0f);
        float w  = S / (S + p + EPSW);
        a[li] = make_float2(z.x * w, z.y * w);
    }
    __syncthreads();

    // inverse FFT along each column (1/1024 applied at store)
    for (int s = 0; s < 10; ++s) {
        for (int q = 0; q < 8; ++q) {
            int tt  = tid + (q << 8);
            int col = tt >> 9;
            int t   = tt & 511;
            fft_stage(a + (col << 10), b + (col << 10), tw, t, s, -1.0f);
        }
        __syncthreads();
        float2* t = a; a = b; b = t;
    }

    for (int li = tid; li < 4096; li += 256) {
        int r = li >> 2, c = li & 3;
        float2 z = a[c * 1024 + r];
        base[(size_t)r * 1024 + c0 + c] = make_float2(z.x * (1.0f / 1024.0f),
                                                      z.y * (1.0f / 1024.0f));
    }
}

// ---------------------------------------------------------------------------
// 5) Inverse 1024-pt FFT per row; write real part (x 1/1024).
// ---------------------------------------------------------------------------
__global__ __launch_bounds__(256) void rowfft_inv(const float2* __restrict__ Z,
                                                  const float2* __restrict__ tw,
                                                  float* __restrict__ out) {
    __shared__ float2 sA[1024];
    __shared__ float2 sB[1024];
    int row = blockIdx.x;
    int tid = threadIdx.x;
    const float2* src = Z + (size_t)row * 1024;
    for (int i = tid; i < 1024; i += 256) sA[i] = src[i];
    __syncthreads();
    float2 *a = sA, *b = sB;
    for (int s = 0; s < 10; ++s) {
        fft_stage(a, b, tw, tid, s, -1.0f);
        fft_stage(a, b, tw, tid + 256, s, -1.0f);
        __syncthreads();
        float2* t = a; a = b; b = t;
    }
    float* dst = out + (size_t)row * 1024;
    for (int i = tid; i < 1024; i += 256) dst[i] = a[i].x * (1.0f / 1024.0f);
}

// ---------------------------------------------------------------------------
// Host-side launcher (graph-capture safe: only kernel launches on `stream`).
// Workspace layout (floats):
//   [0, 2M)          ri    (float2, 1024x1024)
//   [2M, 3M)         psd   (float, 1024x1024)
//   [3M, 3M+32K)     Gre   (32x1024, zero-padded K)
//   [.., +32K)       Gim
//   [.., +2048)      tw    (float2[1024] twiddle table)
//   [.., ..)         Z     (chunk x 1024x1024 float2)
// Images processed in chunks sized from ws_size (deterministic).
// ---------------------------------------------------------------------------
extern "C" void kernel_launch(void* const* d_in, const int* in_sizes, int n_in,
                              void* d_out, int out_size, void* d_ws, size_t ws_size,
                              hipStream_t stream) {
    const float* y   = (const float*)d_in[0];   // [8,3,1024,1024] f32
    const float* psf = (const float*)d_in[1];   // [1,1,25,25]     f32
    float* out = (float*)d_out;
    float* ws  = (float*)d_ws;

    const size_t M1 = 1048576;                  // 1024*1024
    float2* ri  = (float2*)ws;                  // 2M floats
    float*  psd = ws + 2 * M1;                  // 1M floats
    float*  Gre = ws + 3 * M1;                  // 32K floats
    float*  Gim = Gre + 32768;
    float2* tw  = (float2*)(Gim + 32768);       // 2048 floats
    float2* Z   = (float2*)(Gim + 32768 + 2048);

    long long fixed_f = (long long)(3 * M1 + 65536 + 2048);
    long long avail_f = (long long)(ws_size / sizeof(float)) - fixed_f;
    int chunk = (int)(avail_f / (long long)(2 * M1));
    if (chunk < 1)  chunk = 1;
    if (chunk > 24) chunk = 24;

    tw_kernel<<<4, 256, 0, stream>>>(tw);
    psf_g_kernel<<<128, 256, 0, stream>>>(psf, tw, Gre, Gim);
    otf_wmma_kernel<<<4096, 32, 0, stream>>>(Gre, Gim, tw, ri, psd);

    for (int base = 0; base < 24; base += chunk) {
        int nimg = (24 - base) < chunk ? (24 - base) : chunk;
        rowfft_fwd<<<nimg * 1024, 256, 0, stream>>>(y + (size_t)base * M1, tw, Z);
        colfft_wiener<<<nimg * 256, 256, 0, stream>>>(Z, tw, ri, psd);
        rowfft_inv<<<nimg * 1024, 256, 0, stream>>>(Z, tw, out + (size_t)base * M1);
    }
}